// DDH_49246095016535
// MI455X (gfx1250) — compile-verified
//
#include <hip/hip_runtime.h>

// ---------------------------------------------------------------------------
// DDH hashing network forward pass for gfx1250 (MI455X).
// FC layer (256x5820 @ 5820x768) runs on the matrix pipe via
// v_wmma_f32_16x16x4_f32 (exact fp32), with double-buffered LDS tiles filled
// by GLOBAL_LOAD_ASYNC_TO_LDS_B128 (ASYNCcnt-tracked, overlapped with WMMA)
// and deterministic split-K x4 for occupancy. Conv/BN/pool/LC layers are
// fused VALU kernels; every BN is split into a stats-reduction kernel + an
// apply kernel because the reference normalizes with full-batch statistics.
// ---------------------------------------------------------------------------

typedef __attribute__((ext_vector_type(2))) float v2f;
typedef __attribute__((ext_vector_type(8))) float v8f;
typedef __attribute__((__vector_size__(16))) int v4i_t;
typedef __attribute__((address_space(1))) v4i_t* gl_v4i;
typedef __attribute__((address_space(3))) v4i_t* lds_v4i;

constexpr int BATCH = 256;
constexpr int FC_K  = 5820;
constexpr int FC_N  = 768;
constexpr int KSEG  = 1472;   // 46 chunks of 32; 4 segments cover 5820
constexpr int NSEG  = 4;

// ---------------------------------------------------------------------------
// CDNA5 async LDS staging helpers (guarded: fall back to load+store if the
// toolchain does not declare the async builtins). The builtin takes a
// global (AS1) pointer to a 16-byte vector and an LDS (AS3) pointer.
// ---------------------------------------------------------------------------
__device__ __forceinline__ void async_copy_b128(float* lds_dst, const float* gsrc) {
#if __has_builtin(__builtin_amdgcn_global_load_async_to_lds_b128)
  __builtin_amdgcn_global_load_async_to_lds_b128(
      (gl_v4i)gsrc, (lds_v4i)lds_dst, /*offset=*/0, /*cpol=*/0);
#else
  const float4 v = *(const float4*)gsrc;
  *(float4*)lds_dst = v;
#endif
}

__device__ __forceinline__ void wait_async0() {
#if __has_builtin(__builtin_amdgcn_s_wait_asynccnt)
  __builtin_amdgcn_s_wait_asynccnt(0);
#elif __has_builtin(__builtin_amdgcn_global_load_async_to_lds_b128)
  asm volatile("s_wait_asynccnt 0" ::: "memory");
#endif
}

// ---------------------------------------------------------------------------
// Direct convolution, VALID padding, stride 1. One thread per output element.
// ---------------------------------------------------------------------------
template <int Cin, int Cout, int K, int Hin, int Win>
__global__ __launch_bounds__(256) void conv2d_kernel(
    const float* __restrict__ x, const float* __restrict__ w,
    const float* __restrict__ bias, float* __restrict__ out) {
  constexpr int Hout = Hin - K + 1;
  constexpr int Wout = Win - K + 1;
  const int total = BATCH * Cout * Hout * Wout;
  int idx = blockIdx.x * blockDim.x + threadIdx.x;
  if (idx >= total) return;
  int wo = idx % Wout; int t = idx / Wout;
  int ho = t % Hout;   t /= Hout;
  int co = t % Cout;   int n = t / Cout;

  const float* xb = x + (size_t)n * Cin * Hin * Win;
  const float* wc = w + (size_t)co * Cin * K * K;
  float acc = bias[co];
  for (int ci = 0; ci < Cin; ++ci) {
    const float* xc = xb + (size_t)ci * Hin * Win + ho * Win + wo;
    const float* wk = wc + ci * K * K;
#pragma unroll
    for (int kh = 0; kh < K; ++kh)
#pragma unroll
      for (int kw = 0; kw < K; ++kw)
        acc = fmaf(xc[kh * Win + kw], wk[kh * K + kw], acc);
  }
  out[idx] = acc;
}

// ---------------------------------------------------------------------------
// Per-channel batch statistics: stats[2c] = mean, stats[2c+1] = rsqrt(var+eps)
// Input layout [BATCH, C, HW]. One block per channel.
// ---------------------------------------------------------------------------
__global__ __launch_bounds__(256) void bn_stats_kernel(
    const float* __restrict__ x, int C, int HW, float* __restrict__ stats) {
  const int c = blockIdx.x;
  const int tid = threadIdx.x;
  const int count = BATCH * HW;
  float s = 0.f, q = 0.f;
  for (int i = tid; i < count; i += 256) {
    int n = i / HW;
    int p = i - n * HW;
    float v = x[((size_t)n * C + c) * HW + p];
    s += v;
    q += v * v;
  }
  __shared__ float ss[256];
  __shared__ float sq[256];
  ss[tid] = s; sq[tid] = q;
  __syncthreads();
  for (int off = 128; off > 0; off >>= 1) {
    if (tid < off) { ss[tid] += ss[tid + off]; sq[tid] += sq[tid + off]; }
    __syncthreads();
  }
  if (tid == 0) {
    float m = ss[0] / (float)count;
    float var = sq[0] / (float)count - m * m;
    stats[2 * c]     = m;
    stats[2 * c + 1] = rsqrtf(var + 1e-5f);
  }
}

// ---------------------------------------------------------------------------
// Fused BN -> ReLU -> 2x2 maxpool (stride 2). One thread per pooled output.
// ---------------------------------------------------------------------------
__global__ __launch_bounds__(256) void bn_relu_pool_kernel(
    const float* __restrict__ x, const float* __restrict__ stats,
    const float* __restrict__ g, const float* __restrict__ b,
    int C, int Hi, int Wi, float* __restrict__ out) {
  const int Ho = Hi / 2, Wo = Wi / 2;
  const int total = BATCH * C * Ho * Wo;
  int idx = blockIdx.x * blockDim.x + threadIdx.x;
  if (idx >= total) return;
  int wo = idx % Wo; int t = idx / Wo;
  int ho = t % Ho;   t /= Ho;
  int c = t % C;     int n = t / C;
  float scale = g[c] * stats[2 * c + 1];
  float shift = b[c] - stats[2 * c] * scale;
  const float* p = x + ((size_t)n * C + c) * Hi * Wi + (2 * ho) * Wi + 2 * wo;
  float v0 = fmaxf(p[0] * scale + shift, 0.f);
  float v1 = fmaxf(p[1] * scale + shift, 0.f);
  float v2 = fmaxf(p[Wi] * scale + shift, 0.f);
  float v3 = fmaxf(p[Wi + 1] * scale + shift, 0.f);
  out[idx] = fmaxf(fmaxf(v0, v1), fmaxf(v2, v3));
}

// ---------------------------------------------------------------------------
// x1 half of feat: copy h [B,60,7,7] -> feat[:, 0:2940]
// ---------------------------------------------------------------------------
__global__ __launch_bounds__(256) void copy_x1_kernel(
    const float* __restrict__ h, float* __restrict__ feat) {
  const int total = BATCH * 2940;
  int idx = blockIdx.x * blockDim.x + threadIdx.x;
  if (idx >= total) return;
  int n = idx / 2940;
  int i = idx - n * 2940;
  feat[(size_t)n * FC_K + i] = h[idx];
}

// ---------------------------------------------------------------------------
// LocallyConnected2d k=2, s=1 on h [B,60,7,7] -> lc [B,80,6,6]
// lc_w flat layout [80,60,6,6,4] (k = kh*2+kw), lc_b flat [80,36]
// ---------------------------------------------------------------------------
__global__ __launch_bounds__(256) void lc_kernel(
    const float* __restrict__ h, const float* __restrict__ w,
    const float* __restrict__ bias, float* __restrict__ out) {
  const int total = BATCH * 80 * 36;
  int idx = blockIdx.x * blockDim.x + threadIdx.x;
  if (idx >= total) return;
  int j = idx % 6; int t = idx / 6;
  int i = t % 6;   t /= 6;
  int o = t % 80;  int n = t / 80;
  float acc = bias[o * 36 + i * 6 + j];
  const float* hb = h + (size_t)n * 60 * 49 + i * 7 + j;
  const float* wb = w + ((size_t)o * 60 * 36 + i * 6 + j) * 4;
  for (int c = 0; c < 60; ++c) {
    const float* hp = hb + (size_t)c * 49;
    const float* wp = wb + (size_t)c * 36 * 4;
    acc = fmaf(hp[0], wp[0], acc);  // (kh=0,kw=0)
    acc = fmaf(hp[1], wp[1], acc);  // (0,1)
    acc = fmaf(hp[7], wp[2], acc);  // (1,0)
    acc = fmaf(hp[8], wp[3], acc);  // (1,1)
  }
  out[idx] = acc;
}

// ---------------------------------------------------------------------------
// BN -> ReLU on lc [B,80,36], written into feat[:, 2940:5820]
// ---------------------------------------------------------------------------
__global__ __launch_bounds__(256) void bn_relu_to_feat_kernel(
    const float* __restrict__ lc, const float* __restrict__ stats,
    const float* __restrict__ g, const float* __restrict__ b,
    float* __restrict__ feat) {
  const int total = BATCH * 2880;
  int idx = blockIdx.x * blockDim.x + threadIdx.x;
  if (idx >= total) return;
  int n = idx / 2880;
  int i = idx - n * 2880;
  int c = i / 36;
  float scale = g[c] * stats[2 * c + 1];
  float shift = b[c] - stats[2 * c] * scale;
  float v = fmaxf(lc[idx] * scale + shift, 0.f);
  feat[(size_t)n * FC_K + 2940 + i] = v;
}

// ---------------------------------------------------------------------------
// FC GEMM partials: P[seg][256][768] = feat[:, seg] @ fc_w[:, seg]^T
// Exact fp32 on the matrix pipe: v_wmma_f32_16x16x4_f32.
// Block = 256 threads (8 wave32) -> 64x64 output tile; split-K x4 via
// blockIdx.z (segments of 1472 = 46*32 so only the last segment has a tail).
// K is staged 32-wide into double-buffered LDS (row stride 36 floats:
// conflict-free AND keeps every b128 destination 16B-aligned) filled with
// GLOBAL_LOAD_ASYNC_TO_LDS_B128; the async transfer of chunk c+1 overlaps
// the 16 WMMAs of chunk c, then s_wait_asynccnt 0 + barrier swap buffers.
// ---------------------------------------------------------------------------
__global__ __launch_bounds__(256) void fc_wmma_kernel(
    const float* __restrict__ Afeat, const float* __restrict__ Wfc,
    float* __restrict__ P) {
  __shared__ float As[2][64][36];
  __shared__ float Bs[2][64][36];

  const int tid  = threadIdx.x;
  const int lane = tid & 31;
  const int wv   = tid >> 5;        // wave 0..7
  const int msub  = wv & 3;         // 16-row slab within 64-row tile
  const int npair = wv >> 2;        // which 32-col half of 64-col tile
  const int m_base = blockIdx.y * 64;
  const int n_base = blockIdx.x * 64;

  // cooperative-load mapping: 64 rows x 32 k-cols, 8 floats (2 x b128)/thread
  const int lrow  = tid >> 2;       // 0..63
  const int lcol0 = (tid & 3) * 8;  // 0,8,16,24

  // WMMA fragment addressing (ISA 16x16x4 f32 layouts)
  const int arow  = msub * 16 + (lane & 15);
  const int khalf = (lane >> 4) << 1;          // 0 or 2
  const int brow0 = npair * 32 + (lane & 15);

  const int kbeg  = blockIdx.z * KSEG;
  const int kend  = (kbeg + KSEG < FC_K) ? (kbeg + KSEG) : FC_K;
  const int nfull = (kend - kbeg) >> 5;        // full 32-wide chunks
  const int ktail = kbeg + (nfull << 5);       // == kend except last segment

  const float* ga = Afeat + (size_t)(m_base + lrow) * FC_K + lcol0;
  const float* gb = Wfc   + (size_t)(n_base + lrow) * FC_K + lcol0;

  v8f acc0 = {};
  v8f acc1 = {};

  auto issue = [&](int k0, int buf) {
    async_copy_b128(&As[buf][lrow][lcol0],     ga + k0);
    async_copy_b128(&As[buf][lrow][lcol0 + 4], ga + k0 + 4);
    async_copy_b128(&Bs[buf][lrow][lcol0],     gb + k0);
    async_copy_b128(&Bs[buf][lrow][lcol0 + 4], gb + k0 + 4);
  };
  auto compute = [&](int buf) {
#pragma unroll
    for (int kk = 0; kk < 32; kk += 4) {
      v2f a, b0, b1;
      a.x  = As[buf][arow][kk + khalf];
      a.y  = As[buf][arow][kk + khalf + 1];
      b0.x = Bs[buf][brow0][kk + khalf];
      b0.y = Bs[buf][brow0][kk + khalf + 1];
      b1.x = Bs[buf][brow0 + 16][kk + khalf];
      b1.y = Bs[buf][brow0 + 16][kk + khalf + 1];
      acc0 = __builtin_amdgcn_wmma_f32_16x16x4_f32(
          false, a, false, b0, (short)0, acc0, false, false);
      acc1 = __builtin_amdgcn_wmma_f32_16x16x4_f32(
          false, a, false, b1, (short)0, acc1, false, false);
    }
  };

  // prologue: stage chunk 0
  issue(kbeg, 0);
  wait_async0();
  __syncthreads();

  // pipelined main loop: transfer chunk c+1 overlaps WMMAs of chunk c
  for (int c = 0; c < nfull; ++c) {
    const int cur = c & 1;
    if (c + 1 < nfull) issue(kbeg + ((c + 1) << 5), cur ^ 1);
    compute(cur);
    wait_async0();
    __syncthreads();
  }

  // tail chunk (28 valid k-columns in the last segment), zero-padded so the
  // WMMA loop stays uniform with EXEC all-ones
  if (ktail < kend) {
    const int buf = nfull & 1;
#pragma unroll
    for (int u = 0; u < 8; ++u) {
      const int kk = lcol0 + u;
      const int kg = ktail + kk;
      float av = 0.f, bv = 0.f;
      if (kg < kend) {
        av = Afeat[(size_t)(m_base + lrow) * FC_K + kg];
        bv = Wfc[(size_t)(n_base + lrow) * FC_K + kg];
      }
      As[buf][lrow][kk] = av;
      Bs[buf][lrow][kk] = bv;
    }
    __syncthreads();
    compute(buf);
  }

  // write raw partial sums for this K segment (bias added in the reduction)
  const int col0 = n_base + npair * 32 + (lane & 15);
  const int rowb = m_base + msub * 16 + ((lane >> 4) << 3);
  float* Pseg = P + (size_t)blockIdx.z * BATCH * FC_N;
#pragma unroll
  for (int r = 0; r < 8; ++r) {
    Pseg[(size_t)(rowb + r) * FC_N + col0]      = acc0[r];
    Pseg[(size_t)(rowb + r) * FC_N + col0 + 16] = acc1[r];
  }
}

// ---------------------------------------------------------------------------
// Deterministic split-K reduction: Y = bias + sum of the 4 partials (fixed
// order -> bitwise-identical result on every call).
// ---------------------------------------------------------------------------
__global__ __launch_bounds__(256) void fc_reduce_kernel(
    const float* __restrict__ P, const float* __restrict__ bias,
    float* __restrict__ Y) {
  const int total = BATCH * FC_N;
  int idx = blockIdx.x * blockDim.x + threadIdx.x;
  if (idx >= total) return;
  int col = idx % FC_N;
  float s = bias[col];
#pragma unroll
  for (int seg = 0; seg < NSEG; ++seg)
    s += P[(size_t)seg * total + idx];
  Y[idx] = s;
}

// ---------------------------------------------------------------------------
// Final: BN1d(y) -> ReLU -> per-hash fuse dot (16-wide) + bias -> out[256,48]
// ---------------------------------------------------------------------------
__global__ __launch_bounds__(256) void fuse_kernel(
    const float* __restrict__ y, const float* __restrict__ stats,
    const float* __restrict__ g, const float* __restrict__ b,
    const float* __restrict__ fw, const float* __restrict__ fb,
    float* __restrict__ out) {
  const int total = BATCH * 48;
  int idx = blockIdx.x * blockDim.x + threadIdx.x;
  if (idx >= total) return;
  int hh = idx % 48;
  int n = idx / 48;
  float acc = fb[hh];
#pragma unroll
  for (int s = 0; s < 16; ++s) {
    int c = hh * 16 + s;
    float scale = g[c] * stats[2 * c + 1];
    float shift = b[c] - stats[2 * c] * scale;
    float v = fmaxf(y[(size_t)n * FC_N + c] * scale + shift, 0.f);
    acc = fmaf(v, fw[hh * 16 + s], acc);
  }
  out[idx] = acc;
}

// ---------------------------------------------------------------------------
static inline int cdiv_host(int a, int b) { return (a + b - 1) / b; }

extern "C" void kernel_launch(void* const* d_in, const int* in_sizes, int n_in,
                              void* d_out, int out_size, void* d_ws, size_t ws_size,
                              hipStream_t stream) {
  (void)in_sizes; (void)n_in; (void)out_size; (void)ws_size;
  const float* x       = (const float*)d_in[0];
  const float* conv1_w = (const float*)d_in[1];
  const float* conv1_b = (const float*)d_in[2];
  const float* bn1_g   = (const float*)d_in[3];
  const float* bn1_b   = (const float*)d_in[4];
  const float* conv2_w = (const float*)d_in[5];
  const float* conv2_b = (const float*)d_in[6];
  const float* bn2_g   = (const float*)d_in[7];
  const float* bn2_b   = (const float*)d_in[8];
  const float* conv3_w = (const float*)d_in[9];
  const float* conv3_b = (const float*)d_in[10];
  const float* bn3_g   = (const float*)d_in[11];
  const float* bn3_b   = (const float*)d_in[12];
  const float* lc_w    = (const float*)d_in[13];
  const float* lc_b    = (const float*)d_in[14];
  const float* bn4_g   = (const float*)d_in[15];
  const float* bn4_b   = (const float*)d_in[16];
  const float* fc_w    = (const float*)d_in[17];
  const float* fc_b    = (const float*)d_in[18];
  const float* bn5_g   = (const float*)d_in[19];
  const float* bn5_b   = (const float*)d_in[20];
  const float* fuse_w  = (const float*)d_in[21];
  const float* fuse_b  = (const float*)d_in[22];
  float* out = (float*)d_out;

  // Workspace layout (floats). Region A is reused by all three conv outputs.
  float* ws = (float*)d_ws;
  float* convA = ws;                         // max 256*20*62*62 = 19,681,280
  float* pool1 = convA + 19681280;           // 256*20*31*31 = 4,920,320
  float* pool2 = pool1 + 4920320;            // 256*40*15*15 = 2,304,000
  float* hbuf  = pool2 + 2304000;            // 256*60*7*7   =   752,640
  float* lcbuf = hbuf + 752640;              // 256*80*6*6   =   737,280
  float* feat  = lcbuf + 737280;             // 256*5820     = 1,489,920
  float* part  = feat + 1489920;             // 4*256*768    =   786,432
  float* ybuf  = part + 786432;              // 256*768      =   196,608
  float* stats = ybuf + 196608;              // 2*768 (reused per BN layer)

  // ---- layer 1: conv 3->20 (3x3), bn, relu, pool ----
  conv2d_kernel<3, 20, 3, 64, 64>
      <<<cdiv_host(256 * 20 * 62 * 62, 256), 256, 0, stream>>>(x, conv1_w, conv1_b, convA);
  bn_stats_kernel<<<20, 256, 0, stream>>>(convA, 20, 62 * 62, stats);
  bn_relu_pool_kernel<<<cdiv_host(256 * 20 * 31 * 31, 256), 256, 0, stream>>>(
      convA, stats, bn1_g, bn1_b, 20, 62, 62, pool1);

  // ---- layer 2: conv 20->40 (2x2), bn, relu, pool ----
  conv2d_kernel<20, 40, 2, 31, 31>
      <<<cdiv_host(256 * 40 * 30 * 30, 256), 256, 0, stream>>>(pool1, conv2_w, conv2_b, convA);
  bn_stats_kernel<<<40, 256, 0, stream>>>(convA, 40, 30 * 30, stats);
  bn_relu_pool_kernel<<<cdiv_host(256 * 40 * 15 * 15, 256), 256, 0, stream>>>(
      convA, stats, bn2_g, bn2_b, 40, 30, 30, pool2);

  // ---- layer 3: conv 40->60 (2x2), bn, relu, pool ----
  conv2d_kernel<40, 60, 2, 15, 15>
      <<<cdiv_host(256 * 60 * 14 * 14, 256), 256, 0, stream>>>(pool2, conv3_w, conv3_b, convA);
  bn_stats_kernel<<<60, 256, 0, stream>>>(convA, 60, 14 * 14, stats);
  bn_relu_pool_kernel<<<cdiv_host(256 * 60 * 7 * 7, 256), 256, 0, stream>>>(
      convA, stats, bn3_g, bn3_b, 60, 14, 14, hbuf);

  // ---- feat[:, :2940] = h flat ----
  copy_x1_kernel<<<cdiv_host(256 * 2940, 256), 256, 0, stream>>>(hbuf, feat);

  // ---- locally-connected + bn + relu -> feat[:, 2940:] ----
  lc_kernel<<<cdiv_host(256 * 80 * 36, 256), 256, 0, stream>>>(hbuf, lc_w, lc_b, lcbuf);
  bn_stats_kernel<<<80, 256, 0, stream>>>(lcbuf, 80, 36, stats);
  bn_relu_to_feat_kernel<<<cdiv_host(256 * 2880, 256), 256, 0, stream>>>(
      lcbuf, stats, bn4_g, bn4_b, feat);

  // ---- FC GEMM on WMMA (async LDS pipeline, split-K x4) + reduction ----
  fc_wmma_kernel<<<dim3(FC_N / 64, BATCH / 64, NSEG), 256, 0, stream>>>(feat, fc_w, part);
  fc_reduce_kernel<<<cdiv_host(256 * FC_N, 256), 256, 0, stream>>>(part, fc_b, ybuf);

  // ---- bn1d + relu + fuse ----
  bn_stats_kernel<<<FC_N, 256, 0, stream>>>(ybuf, FC_N, 1, stats);
  fuse_kernel<<<cdiv_host(256 * 48, 256), 256, 0, stream>>>(
      ybuf, stats, bn5_g, bn5_b, fuse_w, fuse_b, out);
}